// TensorSquare_80736795230263
// MI455X (gfx1250) — compile-verified
//
#include <hip/hip_runtime.h>
#include <math.h>

// ---------------------------------------------------------------------------
// TensorSquare('64x0e+64x1o+64x2e'), component norm, 512 nodes.
// Output 340MB -> store-bandwidth bound (~14.6us @ 23.3TB/s). f32 end to end.
//  K1: uvuv paths via v_wmma_f32_16x16x4_f32, LDS-staged, async b128 flush.
//  K2: uvu<v / uuu paths via VALU, LDS-staged, async b128 flush.
//  K0: device-side Wigner-3j generation (double precision, exact port).
// ---------------------------------------------------------------------------

#define NODE_OUT 166176
#define IN_DIM   576

#if __has_builtin(__builtin_amdgcn_global_store_async_from_lds_b128)
#define HAVE_ASYNC_STORE 1
typedef __attribute__((ext_vector_type(4))) int v4i_t;
typedef __attribute__((address_space(1))) v4i_t* g_v4i_p;   // global int4*
typedef __attribute__((address_space(3))) v4i_t* l_v4i_p;   // LDS int4*
#else
#define HAVE_ASYNC_STORE 0
#endif

__device__ __forceinline__ void flush16(float* gdst, float* lsrc) {
#if HAVE_ASYNC_STORE
  __builtin_amdgcn_global_store_async_from_lds_b128(
      (g_v4i_p)(void*)gdst, (l_v4i_p)(void*)lsrc, 0, 0);
#else
  *(float4*)gdst = *(const float4*)lsrc;
#endif
}
__device__ __forceinline__ void flush_fence() {
#if HAVE_ASYNC_STORE
  asm volatile("s_wait_asynccnt 0x0" ::: "memory");
#endif
}

// ---- Wigner 3j storage (recomputed every launch; no cross-call reliance) ----
__device__ float g_w3j[1024];

// triples (l1,l2,l3) and their float offsets into g_w3j
__constant__ int c_tri[14][4] = {
  {0,0,0,   0}, {0,1,1,   1}, {0,2,2,  10}, {1,1,0,  35}, {1,1,1,  44},
  {1,1,2,  71}, {1,2,1, 116}, {1,2,2, 161}, {1,2,3, 236}, {2,2,0, 341},
  {2,2,1, 366}, {2,2,2, 441}, {2,2,3, 566}, {2,2,4, 741}
};

struct cplx { double re, im; };
__device__ inline cplx cmul(cplx a, cplx b){ return cplx{a.re*b.re - a.im*b.im, a.re*b.im + a.im*b.re}; }
__device__ inline cplx cconj(cplx a){ return cplx{a.re, -a.im}; }

__device__ double dfact(int n) {            // n <= 9 here: exact in double
  double r = 1.0;
  for (int i = 2; i <= n; ++i) r *= (double)i;
  return r;
}

// faithful port of reference _su2_cg
__device__ double su2_cg(int j1,int m1,int j2,int m2,int j3,int m3) {
  if (m3 != m1 + m2) return 0.0;
  int vmin = 0;
  if (-j1 + j2 + m3 > vmin) vmin = -j1 + j2 + m3;
  if (-j1 + m1     > vmin) vmin = -j1 + m1;
  int vmax = j2 + j3 + m1;
  if (j3 - j1 + j2 < vmax) vmax = j3 - j1 + j2;
  if (j3 + m3      < vmax) vmax = j3 + m3;
  double num = dfact(j3+j1-j2)*dfact(j3-j1+j2)*dfact(j1+j2-j3)*dfact(j3+m3)*dfact(j3-m3);
  double den = dfact(j1+j2+j3+1)*dfact(j1-m1)*dfact(j1+m1)*dfact(j2-m2)*dfact(j2+m2);
  double C = sqrt((2.0*j3 + 1.0) * num / den);
  double S = 0.0;
  for (int v = vmin; v <= vmax; ++v) {
    double t = dfact(j2+j3+m1-v)*dfact(j1-m1+v) /
               (dfact(v)*dfact(j3-j1+j2-v)*dfact(j3+m3-v)*dfact(v+j1-j2-m3));
    S += ((v + j2 + m2) & 1) ? -t : t;
  }
  return C * S;
}

// q_real_to_complex (includes (-i)^l phase)
__device__ void build_q(int l, cplx* q) {
  int d = 2*l + 1;
  for (int a = 0; a < d*d; ++a) q[a] = cplx{0.0, 0.0};
  const double s = 0.7071067811865476;
  for (int m = -l; m < 0; ++m) {
    q[(l+m)*d + (l-m)] = cplx{s, 0.0};
    q[(l+m)*d + (l+m)] = cplx{0.0, -s};
  }
  q[l*d + l] = cplx{1.0, 0.0};
  for (int m = 1; m <= l; ++m) {
    double sg = (m & 1) ? -1.0 : 1.0;
    q[(l+m)*d + (l+m)] = cplx{sg*s, 0.0};
    q[(l+m)*d + (l-m)] = cplx{0.0, sg*s};
  }
  int ph = l & 3;                      // multiply all entries by (-i)^l
  if (ph) {
    for (int a = 0; a < d*d; ++a) {
      cplx v = q[a], r;
      if      (ph == 1) r = cplx{ v.im, -v.re};
      else if (ph == 2) r = cplx{-v.re, -v.im};
      else              r = cplx{-v.im,  v.re};
      q[a] = r;
    }
  }
}

__global__ __launch_bounds__(256) void w3j_init_kernel() {
  __shared__ double csu2[225];
  __shared__ cplx   q1[25], q2[25], q3[81];
  __shared__ double outv[225];
  __shared__ double scale_s;
  int trip = blockIdx.x;
  int l1 = c_tri[trip][0], l2 = c_tri[trip][1], l3 = c_tri[trip][2], off = c_tri[trip][3];
  int d1 = 2*l1+1, d2 = 2*l2+1, d3 = 2*l3+1;
  int n = d1*d2*d3;
  for (int idx = threadIdx.x; idx < n; idx += 256) {
    int i  = idx / (d2*d3);
    int k  = (idx / d3) % d2;
    int nn = idx % d3;
    csu2[idx] = su2_cg(l1, i-l1, l2, k-l2, l3, nn-l3);
  }
  if (threadIdx.x == 0) build_q(l1, q1);
  if (threadIdx.x == 1) build_q(l2, q2);
  if (threadIdx.x == 2) build_q(l3, q3);
  __syncthreads();
  // C_real[j,l,m] = Re( sum_{i,k,n} Q1[i,j] Q2[k,l] conj(Q3[n,m]) Csu2[i,k,n] )
  for (int idx = threadIdx.x; idx < n; idx += 256) {
    int j  = idx / (d2*d3);
    int ll = (idx / d3) % d2;
    int mm = idx % d3;
    double acc = 0.0;
    for (int i = 0; i < d1; ++i) {
      cplx qa = q1[i*d1 + j];
      for (int k = 0; k < d2; ++k) {
        cplx qb = cmul(qa, q2[k*d2 + ll]);
        for (int nn = 0; nn < d3; ++nn) {
          double cv = csu2[(i*d2 + k)*d3 + nn];
          if (cv != 0.0) {
            cplx t = cmul(qb, cconj(q3[nn*d3 + mm]));
            acc += t.re * cv;
          }
        }
      }
    }
    outv[idx] = acc;
  }
  __syncthreads();
  if (threadIdx.x == 0) {
    double s = 0.0;
    for (int i = 0; i < n; ++i) s += outv[i]*outv[i];
    scale_s = (s > 0.0) ? (1.0 / sqrt(s)) : 0.0;
  }
  __syncthreads();
  for (int idx = threadIdx.x; idx < n; idx += 256)
    g_w3j[off + idx] = (float)(outv[idx] * scale_s);
}

// ---------------------------------------------------------------------------
// Kernel 1: 'uvuv' paths via WMMA f32 16x16x4.
//   out[u,v,k] = c*sum_ij cg[i,j,k] x1[u,i] x2[v,j]
//   = per-k GEMM A(64 x d1, K-pad 4) @ B_k(4 x 64).
// Block = (node, group, 16-row u-stripe). Output region of a block is a
// contiguous 1024*dout-float span -> stage in LDS, flush as b128 async stores.
// ---------------------------------------------------------------------------
typedef __attribute__((ext_vector_type(2))) float v2f;
typedef __attribute__((ext_vector_type(8))) float v8f;

struct UvuvMeta { int x1off, d1, x2off, d2, dout, cgoff, outbase; float c; };
__constant__ UvuvMeta c_uvuv[5] = {
  {  0, 1,  64, 3, 3,   1,   6240, 1.7320508075688772f}, // (0,1)->1o
  {  0, 1, 256, 5, 5,  10,  42912, 2.23606797749979f  }, // (0,2)->2e
  { 64, 3, 256, 5, 3, 116,  18528, 1.7320508075688772f}, // (1,2)->1o
  { 64, 3, 256, 5, 5, 161,  84192, 2.23606797749979f  }, // (1,2)->2o
  { 64, 3, 256, 5, 7, 236, 104672, 2.6457513110645907f}, // (1,2)->3o
};

__global__ __launch_bounds__(256) void uvuv_wmma_kernel(const float* __restrict__ x,
                                                        float* __restrict__ out) {
  const int node = blockIdx.z;
  const UvuvMeta m = c_uvuv[blockIdx.y];
  const int tM = blockIdx.x;                 // u-stripe 0..3
  const int dout = m.dout;

  __shared__ float sx1[64*3];                // A rows [u*d1+i]
  __shared__ float sx2[64*5];                // x2 rows [v*d2+j]
  __shared__ float scg[105];                 // c * cg[i,j,k]
  __shared__ __align__(16) float stage[16*64*7];  // <=28KB output staging

  const float* xr = x + (size_t)node * IN_DIM;
  __builtin_prefetch(xr, 0, 1);              // global_prefetch_b8
  const int tid = threadIdx.x;
  for (int t = tid; t < 64*m.d1; t += 256) sx1[t] = xr[m.x1off + t];
  for (int t = tid; t < 64*m.d2; t += 256) sx2[t] = xr[m.x2off + t];
  for (int t = tid; t < m.d1*m.d2*dout; t += 256) scg[t] = m.c * g_w3j[m.cgoff + t];
  __syncthreads();

  const int wave = tid >> 5, lane = tid & 31;
  const int half = lane >> 4, lr = lane & 15;
  const int i0 = half*2, i1 = i0 + 1;        // A: lanes<16 hold K0,K1; >=16 K2,K3
  // A fragment is fixed for the whole block (u-stripe tM)
  const int M = tM*16 + lr;
  v2f a;
  a.x = (i0 < m.d1) ? sx1[M*m.d1 + i0] : 0.f;
  a.y = (i1 < m.d1) ? sx1[M*m.d1 + i1] : 0.f;

  const int ntile = 4 * dout;                // (tN, k) pairs; waves round-robin
  for (int t = wave; t < ntile; t += 8) {
    const int tN = t / dout;
    const int k  = t - tN*dout;
    const int N  = tN*16 + lr;               // B: lane->N, VGPR/half->K row
    float b0 = 0.f, b1 = 0.f;
    if (i0 < m.d1)
      for (int j = 0; j < m.d2; ++j) b0 += scg[(i0*m.d2 + j)*dout + k] * sx2[N*m.d2 + j];
    if (i1 < m.d1)
      for (int j = 0; j < m.d2; ++j) b1 += scg[(i1*m.d2 + j)*dout + k] * sx2[N*m.d2 + j];
    v2f b; b.x = b0; b.y = b1;
    v8f acc = {};
    acc = __builtin_amdgcn_wmma_f32_16x16x4_f32(false, a, false, b,
                                                (short)0, acc, false, false);
    // D: VGPR r -> local row half*8+r, lane -> column N. Odd dout stride in
    // dwords => the 16 lanes of a half hit 16 distinct LDS banks; halves are 512*dout apart (same banks).
#pragma unroll
    for (int r = 0; r < 8; ++r) {
      const int ul = half*8 + r;
      stage[(ul*64 + N)*dout + k] = acc[r];
    }
  }
  __syncthreads();

  // flush: contiguous 1024*dout floats (16B-aligned span), b128 async stores
  float* gbase = out + (size_t)node*NODE_OUT + m.outbase + (size_t)tM*1024*dout;
  const int nchunk = 256 * dout;             // 16B chunks
  for (int c = tid; c < nchunk; c += 256)
    flush16(gbase + c*4, stage + c*4);
  flush_fence();
}

// ---------------------------------------------------------------------------
// Kernel 2: 'uvu<v' (strict upper-tri pairs) and 'uuu' (diagonal) paths.
// One thread per pair; block's output span is contiguous -> LDS stage + b128.
// ---------------------------------------------------------------------------
struct PairMeta { int xoff, din, dout, cgoff, outbase, diag; float c; };
__constant__ PairMeta c_pair[15] = {
  // ---- uvu<v (alpha = 2lo+1) ----
  {   0, 1, 1,   0,      0, 0, 1.0f},                 // (0,0)->0e
  {  64, 3, 1,  35,   2080, 0, 1.0f},                 // (1,1)->0e
  {  64, 3, 3,  44,  30816, 0, 1.7320508075688772f},  // (1,1)->1e
  {  64, 3, 5,  71,  63392, 0, 2.23606797749979f},    // (1,1)->2e
  { 256, 5, 1, 341,   4160, 0, 1.0f},                 // (2,2)->0e
  { 256, 5, 3, 366,  36864, 0, 1.7320508075688772f},  // (2,2)->1e
  { 256, 5, 5, 441,  73792, 0, 2.23606797749979f},    // (2,2)->2e
  { 256, 5, 7, 566, 133344, 0, 2.6457513110645907f},  // (2,2)->3e
  { 256, 5, 9, 741, 147456, 0, 3.0f},                 // (2,2)->4e
  // ---- uuu (diagonal) ----
  {   0, 1, 1,   0,   2016, 1, 0.5773502691896258f},  // (0,0)->0e  c=sqrt(1/3)
  {  64, 3, 1,  35,   4096, 1, 0.4472135954999579f},  // (1,1)->0e  c=sqrt(1/5)
  {  64, 3, 5,  71,  73472, 1, 1.5811388300841898f},  // (1,1)->2e  c=sqrt(5/2)
  { 256, 5, 1, 341,   6176, 1, 0.3779644730092272f},  // (2,2)->0e  c=sqrt(1/7)
  { 256, 5, 5, 441,  83872, 1, 1.5811388300841898f},  // (2,2)->2e  c=sqrt(5/2)
  { 256, 5, 9, 741, 165600, 1, 2.1213203435596424f},  // (2,2)->4e  c=sqrt(9/2)
};

__global__ __launch_bounds__(256) void pair_kernel(const float* __restrict__ x,
                                                   float* __restrict__ out) {
  const int node = blockIdx.z;
  const PairMeta m = c_pair[blockIdx.y];
  if (m.diag && blockIdx.x != 0) return;     // uniform block-level exit

  __shared__ __align__(16) float stage[256*9];
  const int tid = threadIdx.x;
  const int p = blockIdx.x*256 + tid;
  const int npair = m.diag ? 64 : 2016;
  const bool active = p < npair;

  if (active) {
    int u, v;
    if (m.diag) {
      u = v = p;
    } else {
      // invert row-major strict-upper-tri index: offset(u) = u*(127-u)/2
      int uu = (int)((127.0 - sqrt(127.0*127.0 - 8.0*(double)p)) * 0.5);
      if (uu < 0) uu = 0; if (uu > 62) uu = 62;
      while (uu > 0  && p <  uu*(127-uu)/2)           --uu;
      while (uu < 62 && p >= (uu+1)*(127-(uu+1))/2)   ++uu;
      u = uu;
      v = p - uu*(127-uu)/2 + uu + 1;
    }
    const float* xr = x + (size_t)node * IN_DIM + m.xoff;
    float x1v[5], x2v[5];
    for (int i = 0; i < m.din; ++i) {
      x1v[i] = xr[u*m.din + i];
      x2v[i] = xr[v*m.din + i];
    }
    for (int k = 0; k < m.dout; ++k) {
      float acc = 0.f;
      for (int i = 0; i < m.din; ++i)
        for (int j = 0; j < m.din; ++j)
          acc += g_w3j[m.cgoff + (i*m.din + j)*m.dout + k] * x1v[i] * x2v[j];
      stage[tid*m.dout + k] = acc * m.c;
    }
  }
  __syncthreads();

  // flush valid contiguous span (nvalid in {256,224,64} -> multiple of 4 floats)
  const int blockbase = blockIdx.x * 256;
  const int nvalid = (npair - blockbase < 256) ? (npair - blockbase) : 256;
  const int nfl = nvalid * m.dout;
  float* gbase = out + (size_t)node*NODE_OUT + m.outbase + (size_t)blockbase*m.dout;
  for (int c = tid; c*4 < nfl; c += 256)
    flush16(gbase + c*4, stage + c*4);
  flush_fence();
}

// ---------------------------------------------------------------------------
extern "C" void kernel_launch(void* const* d_in, const int* in_sizes, int n_in,
                              void* d_out, int out_size, void* d_ws, size_t ws_size,
                              hipStream_t stream) {
  (void)n_in; (void)d_ws; (void)ws_size; (void)out_size;
  const float* x = (const float*)d_in[0];
  float* out = (float*)d_out;
  const int nodes = in_sizes[0] / IN_DIM;   // 512

  // 0) regenerate Wigner-3j tables on device (deterministic each call)
  w3j_init_kernel<<<dim3(14), dim3(256), 0, stream>>>();
  // 1) uvuv paths: 4 u-stripes x 5 groups x nodes, WMMA f32 16x16x4
  uvuv_wmma_kernel<<<dim3(4, 5, nodes), dim3(256), 0, stream>>>(x, out);
  // 2) triangular + diagonal paths: 15 paths, up to 2016 pairs each
  pair_kernel<<<dim3(8, 15, nodes), dim3(256), 0, stream>>>(x, out);
}